// EnhancedAIDetector_72507637891535
// MI455X (gfx1250) — compile-verified
//
#include <hip/hip_runtime.h>
#include <hip/hip_bf16.h>
#include <math.h>

// ---------------------------------------------------------------------------
// EnhancedAIDetector for MI455X (gfx1250, wave32).
// DFT-as-GEMM via v_wmma_f32_16x16x4_f32 (224 = 14 * 16 tiles), async
// global->LDS double-buffered twiddle staging (ASYNCcnt pipeline), fused
// magnitude+fftshift, exact radix-select percentiles, WMMA FC layers.
// ---------------------------------------------------------------------------

typedef __attribute__((ext_vector_type(2))) float v2f;
typedef __attribute__((ext_vector_type(8))) float v8f;

#define NN 224
#define NPIX (NN * NN)         // 50176
#define NIMG 256
#define PI_D 3.14159265358979323846

__device__ inline v8f wmma4(v2f a, v2f b, v8f c) {
  // D = A(16x4, f32) * B(4x16, f32) + C(16x16, f32)
  return __builtin_amdgcn_wmma_f32_16x16x4_f32(
      false, a, false, b, (short)0, c, false, false);
}

// ---------------------------------------------------------------------------
// K0: DFT twiddle matrices. F[k,n] = exp(-2*pi*i*k*n/224) = C + i*S,
// C = cos(2*pi*kn/224), S = -sin(2*pi*kn/224). Angle reduced exactly via
// (k*n) mod 224, evaluated in double.
// ---------------------------------------------------------------------------
__global__ void k_init(float* __restrict__ Cm, float* __restrict__ Sm) {
  int p = blockIdx.x * 256 + threadIdx.x;
  if (p >= NPIX) return;
  int k = p / NN, n = p - k * NN;
  int m = (k * n) % NN;
  double th = (2.0 * PI_D * (double)m) / (double)NN;
  Cm[p] = (float)cos(th);
  Sm[p] = (float)(-sin(th));
}

// ---------------------------------------------------------------------------
// K1: grayscale
// ---------------------------------------------------------------------------
__global__ void k_gray(const float* __restrict__ x, float* __restrict__ gray) {
  long long i = (long long)blockIdx.x * 256 + threadIdx.x;
  if (i >= (long long)NIMG * NPIX) return;
  int img = (int)(i / NPIX);
  int p = (int)(i - (long long)img * NPIX);
  const float* base = x + (size_t)img * 3 * NPIX;
  gray[i] = 0.299f * base[p] + 0.587f * base[NPIX + p] + 0.114f * base[2 * NPIX + p];
}

// ---------------------------------------------------------------------------
// K2: stage-1 DFT:  M = G @ F  (M_re = G@C, M_im = G@S).
// One wave computes a 16x32 strip of both planes. The 4x32 B panels of the
// C and S twiddle matrices are staged in LDS by a double-buffered
// global_load_async_to_lds_b128 pipeline: one wave-wide async op moves a
// whole 512 B panel; s_wait_asynccnt <= 2 overlaps chunk i+1's transfer with
// chunk i's 4 WMMAs (in-order completion makes the wait exact).
// ---------------------------------------------------------------------------
__global__ __launch_bounds__(32) void k_dft1(const float* __restrict__ gray,
                                             const float* __restrict__ Cm,
                                             const float* __restrict__ Sm,
                                             float* __restrict__ Mre,
                                             float* __restrict__ Mim) {
  __shared__ float tileC[2][128];   // [buffer][4 rows * 32 cols]
  __shared__ float tileS[2][128];

  const int lane = threadIdx.x;
  const int m = lane & 15;            // A row / B col / D col within tile
  const int kb = (lane >> 4) << 1;    // K sub-offset for this half-wave
  int tile = blockIdx.x;              // 14 mtiles * 7 nstrips = 98
  int img = blockIdx.y;
  int mtile = tile / 7, nstrip = tile - mtile * 7;
  int row0 = mtile * 16, col0 = nstrip * 32;
  const float* G = gray + (size_t)img * NPIX;

  // Async staging lane map: lane -> (panel row 0..3, 4-col group 0..7).
  const int arow = lane >> 3;
  const int acg = (lane & 7) << 2;
  // LDS aperture maps addr[31:0] directly to the DS offset (ISA 10.2), so a
  // 32-bit truncation of the generic pointer is the async op's VDST address.
  const unsigned ldsC0 = (unsigned)(unsigned long long)(uintptr_t)&tileC[0][0];
  const unsigned ldsS0 = (unsigned)(unsigned long long)(uintptr_t)&tileS[0][0];
  const unsigned myOff = (unsigned)((arow * 32 + acg) * 4);

  auto issue = [&](int kr0, int buf) {
    const float* gc = Cm + (kr0 + arow) * NN + col0 + acg;
    const float* gs = Sm + (kr0 + arow) * NN + col0 + acg;
    unsigned lc = ldsC0 + (unsigned)buf * 512u + myOff;
    unsigned ls = ldsS0 + (unsigned)buf * 512u + myOff;
    asm volatile("global_load_async_to_lds_b128 %0, %1, off"
                 :: "v"(lc), "v"(gc) : "memory");
    asm volatile("global_load_async_to_lds_b128 %0, %1, off"
                 :: "v"(ls), "v"(gs) : "memory");
  };

  issue(0, 0);

  v8f cr0 = {}, cr1 = {}, ci0 = {}, ci1 = {};
  for (int k0 = 0; k0 < NN; k0 += 4) {
    const int cur = (k0 >> 2) & 1;
    if (k0 + 4 < NN) {
      issue(k0 + 4, cur ^ 1);                       // prefetch next panel pair
      asm volatile("s_wait_asynccnt 0x2" ::: "memory");  // chunk k0 resident
    } else {
      asm volatile("s_wait_asynccnt 0x0" ::: "memory");  // drain tail
    }
    v2f a;
    a.x = G[(row0 + m) * NN + k0 + kb];
    a.y = G[(row0 + m) * NN + k0 + kb + 1];
    const float* tc = tileC[cur];
    const float* ts = tileS[cur];
    v2f bc0, bc1, bs0, bs1;
    bc0.x = tc[kb * 32 + m];       bc0.y = tc[(kb + 1) * 32 + m];
    bc1.x = tc[kb * 32 + 16 + m];  bc1.y = tc[(kb + 1) * 32 + 16 + m];
    bs0.x = ts[kb * 32 + m];       bs0.y = ts[(kb + 1) * 32 + m];
    bs1.x = ts[kb * 32 + 16 + m];  bs1.y = ts[(kb + 1) * 32 + 16 + m];
    cr0 = wmma4(a, bc0, cr0);
    cr1 = wmma4(a, bc1, cr1);
    ci0 = wmma4(a, bs0, ci0);
    ci1 = wmma4(a, bs1, ci1);
  }
  float* R = Mre + (size_t)img * NPIX;
  float* I = Mim + (size_t)img * NPIX;
  int rbase = row0 + ((lane >> 4) << 3);
  for (int r = 0; r < 8; ++r) {
    R[(rbase + r) * NN + col0 + m] = cr0[r];
    R[(rbase + r) * NN + col0 + 16 + m] = cr1[r];
    I[(rbase + r) * NN + col0 + m] = ci0[r];
    I[(rbase + r) * NN + col0 + 16 + m] = ci1[r];
  }
}

// ---------------------------------------------------------------------------
// K3: stage-2 DFT + |.| + fftshift fused:
//   Y_re = C@Mre - S@Mim ; Y_im = C@Mim + S@Mre ; mag[shift(k1),shift(k2)]=|Y|
// ---------------------------------------------------------------------------
__global__ __launch_bounds__(32) void k_dft2(const float* __restrict__ Cm,
                                             const float* __restrict__ Sm,
                                             const float* __restrict__ Mre,
                                             const float* __restrict__ Mim,
                                             float* __restrict__ mag) {
  const int lane = threadIdx.x;
  const int m = lane & 15;
  const int kb = (lane >> 4) << 1;
  int tile = blockIdx.x;              // 14*14 = 196
  int img = blockIdx.y;
  int mtile = tile / 14, ntile = tile - mtile * 14;
  int row0 = mtile * 16, col0 = ntile * 16;
  const float* R = Mre + (size_t)img * NPIX;
  const float* I = Mim + (size_t)img * NPIX;

  v8f yr = {}, yi = {};
  for (int k0 = 0; k0 < NN; k0 += 4) {
    const int kr = k0 + kb;
    __builtin_prefetch(&R[(kr + 4) * NN + col0 + m], 0, 1);
    v2f ac, as, asn, br, bi;
    ac.x = Cm[(row0 + m) * NN + kr];  ac.y = Cm[(row0 + m) * NN + kr + 1];
    as.x = Sm[(row0 + m) * NN + kr];  as.y = Sm[(row0 + m) * NN + kr + 1];
    asn.x = -as.x; asn.y = -as.y;
    br.x = R[kr * NN + col0 + m];     br.y = R[(kr + 1) * NN + col0 + m];
    bi.x = I[kr * NN + col0 + m];     bi.y = I[(kr + 1) * NN + col0 + m];
    yr = wmma4(ac, br, yr);
    yr = wmma4(asn, bi, yr);
    yi = wmma4(ac, bi, yi);
    yi = wmma4(as, br, yi);
  }
  float* Mg = mag + (size_t)img * NPIX;
  int rbase = row0 + ((lane >> 4) << 3);
  int c = col0 + m;
  int sc = (c >= 112) ? c - 112 : c + 112;
  for (int r = 0; r < 8; ++r) {
    int row = rbase + r;
    int sr = (row >= 112) ? row - 112 : row + 112;
    Mg[sr * NN + sc] = sqrtf(yr[r] * yr[r] + yi[r] * yi[r]);
  }
}

// ---------------------------------------------------------------------------
// Block reductions (work for blockDim 64 and 256)
// ---------------------------------------------------------------------------
__device__ float brSum(float v, float* s) {
  int t = threadIdx.x;
  s[t] = v; __syncthreads();
  for (int off = blockDim.x >> 1; off > 0; off >>= 1) {
    if (t < off) s[t] += s[t + off];
    __syncthreads();
  }
  float r = s[0]; __syncthreads();
  return r;
}
__device__ float brMax(float v, float* s) {
  int t = threadIdx.x;
  s[t] = v; __syncthreads();
  for (int off = blockDim.x >> 1; off > 0; off >>= 1) {
    if (t < off) s[t] = fmaxf(s[t], s[t + off]);
    __syncthreads();
  }
  float r = s[0]; __syncthreads();
  return r;
}
__device__ float brMin(float v, float* s) {
  int t = threadIdx.x;
  s[t] = v; __syncthreads();
  for (int off = blockDim.x >> 1; off > 0; off >>= 1) {
    if (t < off) s[t] = fminf(s[t], s[t + off]);
    __syncthreads();
  }
  float r = s[0]; __syncthreads();
  return r;
}

// ---------------------------------------------------------------------------
// Exact k-th order statistic over {mag[p] : lo2 <= d2(p) < hi2} via MSB-first
// radix select on float bit patterns (values >= 0 => bits order-isomorphic).
// Must be called by all 256 threads of the block.
// ---------------------------------------------------------------------------
__device__ float kthValue(const float* __restrict__ Mg, int lo2, int hi2, int k,
                          unsigned* hist, int* sscal) {
  unsigned prefix = 0u;
  int r = k;
  for (int pass = 0; pass < 4; ++pass) {
    int shift = 24 - 8 * pass;
    hist[threadIdx.x] = 0u;     // blockDim == 256 == #bins
    __syncthreads();
    for (int p = threadIdx.x; p < NPIX; p += 256) {
      int yy = p / NN, xx = p - yy * NN;
      int dy = yy - 112, dx = xx - 112;
      int d2 = dx * dx + dy * dy;
      if (d2 >= lo2 && d2 < hi2) {
        unsigned key = __float_as_uint(Mg[p]);
        bool ok = (pass == 0) || ((key >> (shift + 8)) == (prefix >> (shift + 8)));
        if (ok) atomicAdd(&hist[(key >> shift) & 255u], 1u);
      }
    }
    __syncthreads();
    if (threadIdx.x == 0) {
      int rr = r, b = 0;
      for (; b < 255; ++b) {
        int c = (int)hist[b];
        if (rr < c) break;
        rr -= c;
      }
      sscal[0] = rr; sscal[1] = b;
    }
    __syncthreads();
    r = sscal[0];
    prefix |= ((unsigned)sscal[1]) << shift;
    __syncthreads();
  }
  return __uint_as_float(prefix);
}

// jnp.percentile (linear interp): a[k] + frac*(a[k+1]-a[k])
__device__ float bandPercentile(const float* __restrict__ Mg, int lo2, int hi2,
                                int n, double q, unsigned* hist, int* sscal,
                                float* sred) {
  double pos = (q / 100.0) * (double)(n - 1);
  int k = (int)pos;
  float frac = (float)(pos - (double)k);
  float vlo = kthValue(Mg, lo2, hi2, k, hist, sscal);
  unsigned keylo = __float_as_uint(vlo);
  float cle = 0.f, mgt = 3.4e38f;
  for (int p = threadIdx.x; p < NPIX; p += 256) {
    int yy = p / NN, xx = p - yy * NN;
    int dy = yy - 112, dx = xx - 112;
    int d2 = dx * dx + dy * dy;
    if (d2 >= lo2 && d2 < hi2) {
      float v = Mg[p];
      if (__float_as_uint(v) <= keylo) cle += 1.f;
      else mgt = fminf(mgt, v);
    }
  }
  float cntLe = brSum(cle, sred);
  float minGt = brMin(mgt, sred);
  float vhi = (cntLe >= (float)(k + 2)) ? vlo : minGt;
  return vlo + frac * (vhi - vlo);
}

// ---------------------------------------------------------------------------
// K4: all 24 frequency-domain stats for one image per 256-thread block.
// ---------------------------------------------------------------------------
__global__ __launch_bounds__(256) void k_freq(const float* __restrict__ mag,
                                              float* __restrict__ feat) {
  __shared__ float sred[256];
  __shared__ unsigned hist[256];
  __shared__ int sscal[2];
  __shared__ float sfeat[24];
  const int img = blockIdx.x;
  const int tid = threadIdx.x;
  const float* Mg = mag + (size_t)img * NPIX;

  // ---- fused sweep: bands 0/1/2, full, high ----
  float s[5] = {0, 0, 0, 0, 0}, q2[5] = {0, 0, 0, 0, 0};
  float mx[5] = {-1e30f, -1e30f, -1e30f, -1e30f, -1e30f};
  float cnt[4] = {0, 0, 0, 0};  // b0,b1,b2,high
  for (int p = tid; p < NPIX; p += 256) {
    float v = Mg[p];
    int yy = p / NN, xx = p - yy * NN;
    int dy = yy - 112, dx = xx - 112;
    int d2 = dx * dx + dy * dy;
    s[3] += v; q2[3] += v * v; mx[3] = fmaxf(mx[3], v);
    if (d2 < 400)       { s[0] += v; q2[0] += v * v; mx[0] = fmaxf(mx[0], v); cnt[0] += 1.f; }
    else if (d2 < 2500) { s[1] += v; q2[1] += v * v; mx[1] = fmaxf(mx[1], v); cnt[1] += 1.f; }
    else if (d2 < 10000){ s[2] += v; q2[2] += v * v; mx[2] = fmaxf(mx[2], v); cnt[2] += 1.f; }
    if (d2 > 6400)      { s[4] += v; q2[4] += v * v; mx[4] = fmaxf(mx[4], v); cnt[3] += 1.f; }
  }
  float S[5], Q[5], MX[5], CN[4];
  for (int i = 0; i < 5; ++i) {
    S[i] = brSum(s[i], sred);
    Q[i] = brSum(q2[i], sred);
    MX[i] = brMax(mx[i], sred);
  }
  for (int i = 0; i < 4; ++i) CN[i] = brSum(cnt[i], sred);
  int n0 = (int)(CN[0] + 0.5f), n1 = (int)(CN[1] + 0.5f);
  int n2 = (int)(CN[2] + 0.5f), nh = (int)(CN[3] + 0.5f);

  // ---- center lines ----
  float hlv = 0.f, hlq = 0.f, vlv = 0.f, vlq = 0.f;
  if (tid < NN) {
    float a = Mg[112 * NN + tid];
    float b = Mg[tid * NN + 112];
    hlv = a; hlq = a * a; vlv = b; vlq = b * b;
  }
  float HS = brSum(hlv, sred), HQ = brSum(hlq, sred);
  float VS = brSum(vlv, sred), VQ = brSum(vlq, sred);

  // ---- high-band count(v > 2*mean) ----
  float meanH = S[4] / (float)nh;
  float hc = 0.f;
  for (int p = tid; p < NPIX; p += 256) {
    int yy = p / NN, xx = p - yy * NN;
    int dy = yy - 112, dx = xx - 112;
    if (dx * dx + dy * dy > 6400 && Mg[p] > 2.f * meanH) hc += 1.f;
  }
  float HC = brSum(hc, sred);

  // ---- exact percentiles ----
  float p95b0 = bandPercentile(Mg, 0, 400, n0, 95.0, hist, sscal, sred);
  float p95b1 = bandPercentile(Mg, 400, 2500, n1, 95.0, hist, sscal, sred);
  float p95b2 = bandPercentile(Mg, 2500, 10000, n2, 95.0, hist, sscal, sred);
  float p95f  = bandPercentile(Mg, 0, 0x7FFFFFFF, NPIX, 95.0, hist, sscal, sred);
  float p5f   = bandPercentile(Mg, 0, 0x7FFFFFFF, NPIX, 5.0, hist, sscal, sred);

  if (tid == 0) {
    float nB[3] = {(float)n0, (float)n1, (float)n2};
    float p95B[3] = {p95b0, p95b1, p95b2};
    for (int i = 0; i < 3; ++i) {
      float mean = S[i] / nB[i];
      float stdv = sqrtf(fmaxf(0.f, Q[i] / nB[i] - mean * mean));
      sfeat[4 * i + 0] = mean;
      sfeat[4 * i + 1] = stdv;
      sfeat[4 * i + 2] = MX[i];
      sfeat[4 * i + 3] = p95B[i];
    }
    float fm = S[3] / (float)NPIX;
    sfeat[12] = fm;
    sfeat[13] = sqrtf(fmaxf(0.f, Q[3] / (float)NPIX - fm * fm));
    sfeat[14] = MX[3];
    sfeat[15] = p95f;
    sfeat[16] = p5f;
    float hm = HS / 224.f, vm = VS / 224.f;
    sfeat[17] = hm;
    sfeat[18] = sqrtf(fmaxf(0.f, HQ / 224.f - hm * hm));
    sfeat[19] = vm;
    sfeat[20] = sqrtf(fmaxf(0.f, VQ / 224.f - vm * vm));
    sfeat[21] = meanH;
    sfeat[22] = sqrtf(fmaxf(0.f, Q[4] / (float)nh - meanH * meanH));
    sfeat[23] = HC;
  }
  __syncthreads();
  feat[(size_t)img * 256 + tid] = (tid < 24) ? sfeat[tid] : 0.f;
}

// ---------------------------------------------------------------------------
// K5: 8x8 DCT-II block stats. One 64-thread block per DCT block.
// ---------------------------------------------------------------------------
__global__ __launch_bounds__(64) void k_dct(const float* __restrict__ gray,
                                            float* __restrict__ dctF) {
  __shared__ float X[64], D[64], T[64], Dd[64], red[64];
  int bi = blockIdx.x;   // 0..62
  int img = blockIdx.y;
  int t = threadIdx.x;
  int br = bi / 28, bc = bi - br * 28;
  int a = t >> 3, b = t & 7;
  X[t] = gray[(size_t)img * NPIX + (br * 8 + a) * NN + bc * 8 + b];
  D[t] = cosf((float)PI_D * (float)(2 * b + 1) * (float)a / 16.0f) *
         (a == 0 ? 0.35355339059327373f : 0.5f);
  __syncthreads();
  float acc = 0.f;                       // T[k,c] = sum_a D[k,a]*X[a,c]
  for (int u = 0; u < 8; ++u) acc += D[(t >> 3) * 8 + u] * X[u * 8 + (t & 7)];
  T[t] = acc; __syncthreads();
  acc = 0.f;                             // Dd[k,l] = sum_c T[k,c]*D[l,c]
  for (int u = 0; u < 8; ++u) acc += T[(t >> 3) * 8 + u] * D[(t & 7) * 8 + u];
  Dd[t] = acc; __syncthreads();

  float v = (t >= 1) ? Dd[t] : 0.f;      // AC coefficients only
  float av = fabsf(v);
  float sAbs = brSum(av, red);
  float sV = brSum(v, red);
  float sV2 = brSum(v * v, red);
  float mxA = brMax((t >= 1) ? av : -1e30f, red);
  float meanV = sV / 63.f;
  float stdv = sqrtf(fmaxf(0.f, sV2 / 63.f - meanV * meanV));
  float cntf = brSum((t >= 1 && av > stdv) ? 1.f : 0.f, red);
  if (t == 0) {
    float* o = dctF + (size_t)img * 256 + bi * 4;
    o[0] = sAbs / 63.f; o[1] = stdv; o[2] = mxA; o[3] = cntf;
  }
  if (bi == 0 && t < 4) dctF[(size_t)img * 256 + 252 + t] = 0.f;  // pad tail
}

// ---------------------------------------------------------------------------
// K6: FC layer: out[:, colOff:colOff+512] = feats(256x256) @ W(256x512) + b
// ---------------------------------------------------------------------------
__global__ __launch_bounds__(32) void k_fc(const float* __restrict__ feats,
                                           const float* __restrict__ W,
                                           const float* __restrict__ bias,
                                           float* __restrict__ out, int colOff) {
  const int lane = threadIdx.x;
  const int m = lane & 15;
  const int kb = (lane >> 4) << 1;
  int row0 = blockIdx.x * 16;   // 16 tiles of images
  int col0 = blockIdx.y * 16;   // 32 tiles of features
  v8f acc = {};
  for (int k0 = 0; k0 < 256; k0 += 4) {
    const int kr = k0 + kb;
    v2f a, bvec;
    a.x = feats[(row0 + m) * 256 + kr];
    a.y = feats[(row0 + m) * 256 + kr + 1];
    bvec.x = W[kr * 512 + col0 + m];
    bvec.y = W[(kr + 1) * 512 + col0 + m];
    acc = wmma4(a, bvec, acc);
  }
  int rbase = row0 + ((lane >> 4) << 3);
  float bb = bias[col0 + m];
  for (int r = 0; r < 8; ++r)
    out[(size_t)(rbase + r) * 1024 + colOff + col0 + m] = acc[r] + bb;
}

// ---------------------------------------------------------------------------
extern "C" void kernel_launch(void* const* d_in, const int* in_sizes, int n_in,
                              void* d_out, int out_size, void* d_ws, size_t ws_size,
                              hipStream_t stream) {
  const float* x  = (const float*)d_in[0];
  const float* Wf = (const float*)d_in[1];
  const float* bf = (const float*)d_in[2];
  const float* Wd = (const float*)d_in[3];
  const float* bd = (const float*)d_in[4];
  float* out = (float*)d_out;

  // workspace layout (floats); total ~207 MB
  float* ws   = (float*)d_ws;
  float* Cm   = ws;                       // 50176
  float* Sm   = Cm + NPIX;                // 50176
  float* gray = Sm + NPIX;                // 256*50176
  float* Mre  = gray + (size_t)NIMG * NPIX;
  float* Mim  = Mre + (size_t)NIMG * NPIX;
  float* mag  = Mim + (size_t)NIMG * NPIX;
  float* fftF = mag + (size_t)NIMG * NPIX;   // 256*256
  float* dctF = fftF + NIMG * 256;           // 256*256

  k_init<<<(NPIX + 255) / 256, 256, 0, stream>>>(Cm, Sm);
  k_gray<<<(NIMG * NPIX + 255) / 256, 256, 0, stream>>>(x, gray);
  k_dft1<<<dim3(98, NIMG), 32, 0, stream>>>(gray, Cm, Sm, Mre, Mim);
  k_dft2<<<dim3(196, NIMG), 32, 0, stream>>>(Cm, Sm, Mre, Mim, mag);
  k_freq<<<NIMG, 256, 0, stream>>>(mag, fftF);
  k_dct<<<dim3(63, NIMG), 64, 0, stream>>>(gray, dctF);
  k_fc<<<dim3(16, 32), 32, 0, stream>>>(fftF, Wf, bf, out, 0);
  k_fc<<<dim3(16, 32), 32, 0, stream>>>(dctF, Wd, bd, out, 512);
}